// GCN_79998060855857
// MI455X (gfx1250) — compile-verified
//
#include <hip/hip_runtime.h>

#define N_NODES 100000
#define N_EDGES 1600000
#define C_IN    128
#define C_H     128
#define C_OUT   64
#define BN_EPS  1e-5f

typedef float v2f __attribute__((ext_vector_type(2)));
typedef float v8f __attribute__((ext_vector_type(8)));

// ---------------- degree / normalization ----------------

__global__ void k_init_deg(float* deg) {
    int i = blockIdx.x * blockDim.x + threadIdx.x;
    if (i < N_NODES) deg[i] = 1.0f;               // self-loop contributes 1
}

__global__ void k_deg_accum(const long long* __restrict__ ei, float* deg) {
    int e = blockIdx.x * blockDim.x + threadIdx.x;
    if (e < N_EDGES) atomicAdd(&deg[(int)ei[N_EDGES + e]], 1.0f);
}

__global__ void k_dinv(float* deg) {
    int i = blockIdx.x * blockDim.x + threadIdx.x;
    if (i < N_NODES) deg[i] = rsqrtf(deg[i]);     // deg >= 1 always
}

// ---------------- fp32 WMMA GEMM: H = X @ W  (K = 128) ----------------
// grid: (ceil(rowTiles/8), Cout/16), block: 256 (8 waves, one 16x16 tile each)

__global__ void k_gemm_wmma(const float* __restrict__ X, const float* __restrict__ W,
                            float* __restrict__ H, int rowTiles, int Cout) {
    int lane = threadIdx.x & 31;
    int wv   = threadIdx.x >> 5;
    int tile = blockIdx.x * 8 + wv;
    if (tile >= rowTiles) return;                  // wave-uniform: EXEC all-1s inside
    int row0 = tile * 16;
    int col0 = blockIdx.y * 16;

    int m    = lane & 15;                          // M (for A) / N (for B) index
    int koff = (lane >> 4) << 1;                   // lanes 16-31 hold K+2,K+3

    v8f c = {};
    const float* xrow = X + (size_t)(row0 + m) * C_IN;
    for (int k = 0; k < C_IN; k += 4) {
        int kk = k + koff;
        v2f a, b;
        a[0] = xrow[kk];
        a[1] = xrow[kk + 1];
        const float* wp = W + (size_t)kk * Cout + col0 + m;
        b[0] = wp[0];
        b[1] = wp[Cout];
        c = __builtin_amdgcn_wmma_f32_16x16x4_f32(false, a, false, b,
                                                  (short)0, c, false, false);
    }
    // D layout: VGPR j -> M=j (lanes 0-15), M=8+j (lanes 16-31); N = lane&15
    int rbase = row0 + ((lane >> 4) << 3);
    float* outp = H + (size_t)rbase * Cout + col0 + m;
    #pragma unroll
    for (int j = 0; j < 8; ++j) outp[(size_t)j * Cout] = c[j];
}

// ---------------- self-loop init: agg[i,:] = h[i,:] * dinv[i]^2 ----------------

__global__ void k_selfloop(const float* __restrict__ h, const float* __restrict__ dinv,
                           float* __restrict__ agg, int C) {
    long long idx = (long long)blockIdx.x * blockDim.x + threadIdx.x;
    if (idx < (long long)N_NODES * C) {
        int i = (int)(idx / C);
        float d = dinv[i];
        agg[idx] = h[idx] * d * d;
    }
}

// ---------------- edge scatter: agg[dst,:] += h[src,:] * dinv[src]*dinv[dst] ----

__global__ void k_scatter(const long long* __restrict__ ei, const float* __restrict__ h,
                          const float* __restrict__ dinv, float* __restrict__ agg, int C) {
    int per = C >> 2;                              // float4 groups per edge
    long long idx = (long long)blockIdx.x * blockDim.x + threadIdx.x;
    if (idx >= (long long)N_EDGES * per) return;
    int e  = (int)(idx / per);
    int cg = (int)(idx % per) << 2;
    int s = (int)ei[e];
    int d = (int)ei[N_EDGES + e];
    float w = dinv[s] * dinv[d];
    const float4 hv = *reinterpret_cast<const float4*>(h + (size_t)s * C + cg);
    float* ap = agg + (size_t)d * C + cg;
    atomicAdd(ap + 0, hv.x * w);
    atomicAdd(ap + 1, hv.y * w);
    atomicAdd(ap + 2, hv.z * w);
    atomicAdd(ap + 3, hv.w * w);
}

// ---------------- BatchNorm (batch statistics) ----------------

__global__ void k_zero_stats(float* stats, int n) {
    int i = threadIdx.x;
    if (i < n) stats[i] = 0.0f;
}

// block = C threads (one channel each), each block covers rowsPerBlock rows
__global__ void k_bn_stats(const float* __restrict__ x, float* __restrict__ stats,
                           int rowsPerBlock) {
    int C = blockDim.x;
    int c = threadIdx.x;
    int r0 = blockIdx.x * rowsPerBlock;
    int r1 = r0 + rowsPerBlock; if (r1 > N_NODES) r1 = N_NODES;
    float s = 0.0f, s2 = 0.0f;
    for (int r = r0; r < r1; ++r) {
        float v = x[(size_t)r * C + c];
        s += v; s2 += v * v;
    }
    atomicAdd(&stats[c], s);
    atomicAdd(&stats[C + c], s2);
}

__global__ void k_bn_finalize(float* stats, int C) {
    int c = threadIdx.x;
    if (c < C) {
        float mu  = stats[c] / (float)N_NODES;
        float var = stats[C + c] / (float)N_NODES - mu * mu;
        stats[2 * C + c] = mu;
        stats[3 * C + c] = rsqrtf(var + BN_EPS);
    }
}

__global__ void k_bn_apply(const float* __restrict__ x, const float* __restrict__ stats,
                           const float* __restrict__ g, const float* __restrict__ be,
                           float* __restrict__ y, int C, int relu) {
    long long idx = (long long)blockIdx.x * blockDim.x + threadIdx.x;
    if (idx >= (long long)N_NODES * C) return;
    int c = (int)(idx % C);
    float v = (x[idx] - stats[2 * C + c]) * stats[3 * C + c] * g[c] + be[c];
    if (relu) v = fmaxf(v, 0.0f);
    y[idx] = v;
}

// ---------------- host orchestration ----------------

extern "C" void kernel_launch(void* const* d_in, const int* in_sizes, int n_in,
                              void* d_out, int out_size, void* d_ws, size_t ws_size,
                              hipStream_t stream) {
    const float*     x   = (const float*)d_in[0];
    const long long* ei  = (const long long*)d_in[1];   // reference dtype: int64
    const float* W1 = (const float*)d_in[2];
    const float* g1 = (const float*)d_in[4];
    const float* be1 = (const float*)d_in[5];
    const float* W2 = (const float*)d_in[6];
    const float* g2 = (const float*)d_in[8];
    const float* be2 = (const float*)d_in[9];
    const float* W3 = (const float*)d_in[10];
    const float* g3 = (const float*)d_in[12];
    const float* be3 = (const float*)d_in[13];
    float* out = (float*)d_out;

    // workspace layout (floats): deg[N] | h[N*128] | agg[N*128] | act[N*128] | stats[512]
    float* deg   = (float*)d_ws;
    float* h     = deg + N_NODES;
    float* agg   = h   + (size_t)N_NODES * C_H;
    float* act   = agg + (size_t)N_NODES * C_H;
    float* stats = act + (size_t)N_NODES * C_H;

    // GCN symmetric normalization
    k_init_deg <<<(N_NODES + 255) / 256, 256, 0, stream>>>(deg);
    k_deg_accum<<<(N_EDGES + 255) / 256, 256, 0, stream>>>(ei, deg);
    k_dinv     <<<(N_NODES + 255) / 256, 256, 0, stream>>>(deg);

    const int rowTiles = N_NODES / 16;              // 6250 exactly

    auto layer = [&](const float* in, const float* Wt, const float* gg, const float* bb,
                     float* outbuf, int Cout, int relu) {
        dim3 gg_gemm((rowTiles + 7) / 8, Cout / 16);
        k_gemm_wmma<<<gg_gemm, 256, 0, stream>>>(in, Wt, h, rowTiles, Cout);

        long long nc = (long long)N_NODES * Cout;
        k_selfloop<<<(unsigned)((nc + 255) / 256), 256, 0, stream>>>(h, deg, agg, Cout);

        long long sc = (long long)N_EDGES * (Cout / 4);
        k_scatter<<<(unsigned)((sc + 255) / 256), 256, 0, stream>>>(ei, h, deg, agg, Cout);

        k_zero_stats <<<1, 256, 0, stream>>>(stats, 2 * Cout);
        k_bn_stats   <<<(N_NODES + 255) / 256, Cout, 0, stream>>>(agg, stats, 256);
        k_bn_finalize<<<1, Cout, 0, stream>>>(stats, Cout);
        k_bn_apply   <<<(unsigned)((nc + 255) / 256), 256, 0, stream>>>(
            agg, stats, gg, bb, outbuf, Cout, relu);
    };

    layer(x,   W1, g1, be1, act, C_H,   1);   // bias b* cancels inside BatchNorm
    layer(act, W2, g2, be2, act, C_H,   1);
    layer(act, W3, g3, be3, out, C_OUT, 0);
}